// DeformConvNetV1V2_64974265254232
// MI455X (gfx1250) — compile-verified
//
#include <hip/hip_runtime.h>
#include <hip/hip_bf16.h>

typedef float v2f __attribute__((ext_vector_type(2)));
typedef float v8f __attribute__((ext_vector_type(8)));

#define BN_EPS 1e-5f

// ---------------------------------------------------------------------------
// Stage 1: conv 3x3, 3->32, stride1 pad1, +bias, +BN+ReLU.  Implicit GEMM:
// M = 256*32*32 pixels, K = 27 (pad 28), N = 32.  WMMA f32 16x16x4.
// One wave per 16-pixel M tile; 2 N-tiles of accumulators.
// ---------------------------------------------------------------------------
__global__ __launch_bounds__(128) void conv1_wmma(
    const float* __restrict__ x,    // (256,3,32,32)
    const float* __restrict__ w1,   // (32,3,3,3)
    const float* __restrict__ b1,   // (32)
    const float* __restrict__ g,  const float* __restrict__ be,
    const float* __restrict__ mn, const float* __restrict__ vr,
    float* __restrict__ out)        // (256,32,32,32)
{
    const int C = 3, H = 32, W = 32, Cout = 32;
    __shared__ float Bt[28 * 32];      // B tile: K x N
    __shared__ float At[4][16 * 28];   // per-wave A tile: M x K

    const int tid  = threadIdx.x;
    const int wave = tid >> 5;
    const int lane = tid & 31;
    const int mtile = blockIdx.x * 4 + wave;
    const int b    = (mtile * 16) >> 10;        // /(H*W)
    const int rem0 = (mtile * 16) & 1023;       // %(H*W)

    // Stage weights (K-major, pad k=27 with zeros)
    for (int e = tid; e < 28 * 32; e += 128) {
        int k = e >> 5, o = e & 31;
        Bt[e] = (k < 27) ? w1[o * 27 + k] : 0.0f;
    }
    // Stage im2col A tile for this wave's 16 pixels
    for (int e = lane; e < 16 * 28; e += 32) {
        int m = e / 28, k = e % 28;
        float v = 0.0f;
        if (k < 27) {
            int rem = rem0 + m;
            int y = rem >> 5, xx = rem & 31;
            int c = k / 9, t = k % 9;
            int iy = y + t / 3 - 1, ix = xx + t % 3 - 1;
            if (iy >= 0 && iy < H && ix >= 0 && ix < W)
                v = x[(((long)b * C + c) * H + iy) * W + ix];
        }
        At[wave][m * 28 + k] = v;
    }
    __syncthreads();

    v8f acc0 = {}; v8f acc1 = {};
    const int mrow = lane & 15;            // A row / B column (within tile)
    const int koff = (lane >> 4) << 1;     // upper half-wave holds K+2,K+3
    const float* Ar = &At[wave][mrow * 28];
#pragma unroll
    for (int s = 0; s < 7; ++s) {
        int kb = s * 4 + koff;
        v2f a;  a.x  = Ar[kb];               a.y  = Ar[kb + 1];
        v2f b0; b0.x = Bt[kb * 32 + mrow];   b0.y = Bt[(kb + 1) * 32 + mrow];
        v2f b1v; b1v.x = Bt[kb * 32 + mrow + 16]; b1v.y = Bt[(kb + 1) * 32 + mrow + 16];
        acc0 = __builtin_amdgcn_wmma_f32_16x16x4_f32(false, a, false, b0,  (short)0, acc0, false, false);
        acc1 = __builtin_amdgcn_wmma_f32_16x16x4_f32(false, a, false, b1v, (short)0, acc1, false, false);
    }

    // Epilogue: (+bias) then BN + ReLU, scatter per documented C/D layout
    const int mb = (lane >> 4) * 8;
#pragma unroll
    for (int nt = 0; nt < 2; ++nt) {
        int o = mrow + nt * 16;
        float sc = g[o] * rsqrtf(vr[o] + BN_EPS);
        float sh = be[o] - mn[o] * sc + b1[o] * sc;
        v8f a = nt ? acc1 : acc0;
#pragma unroll
        for (int r = 0; r < 8; ++r) {
            int rem = rem0 + mb + r;
            int y = rem >> 5, xx = rem & 31;
            float v = a[r] * sc + sh;
            out[(((long)b * Cout + o) * H + y) * W + xx] = v > 0.f ? v : 0.f;
        }
    }
}

// ---------------------------------------------------------------------------
// Offset conv: 3x3, pad1, stride s, Cin -> 18, +bias.  Small N; direct VALU.
// Fully templated geometry -> all div/mod fold to shifts/magic constants.
// ---------------------------------------------------------------------------
template <int C, int H, int W, int h, int w, int STRIDE>
__global__ void offset_conv(
    const float* __restrict__ x, const float* __restrict__ wp,
    const float* __restrict__ bp, float* __restrict__ out)
{
    const int idx = blockIdx.x * blockDim.x + threadIdx.x;
    if (idx >= 256 * 18 * h * w) return;
    int xx = idx & (w - 1); int t = idx / w;
    int y = t & (h - 1); t /= h;
    int o = t % 18; int b = t / 18;
    float acc = bp[o];
    int iy0 = y * STRIDE - 1, ix0 = xx * STRIDE - 1;
    for (int c = 0; c < C; ++c) {
        const float* xp  = x + ((long)b * C + c) * H * W;
        const float* wpp = wp + ((long)o * C + c) * 9;
#pragma unroll
        for (int ky = 0; ky < 3; ++ky) {
            int iy = iy0 + ky;
            if (iy < 0 || iy >= H) continue;
#pragma unroll
            for (int kx = 0; kx < 3; ++kx) {
                int ix = ix0 + kx;
                if (ix < 0 || ix >= W) continue;
                acc += xp[iy * W + ix] * wpp[ky * 3 + kx];
            }
        }
    }
    out[idx] = acc;
}

// ---------------------------------------------------------------------------
// Fused deformable sampling + GEMM (+BN+ReLU).  GEMM: M = B*h*w pixels,
// K = C*9, N = Cout.  Per wave: 16-pixel M tile, NT accumulator tiles.
// K processed in chunks of 36 (4 channels x 9 taps) = 9 WMMA k-steps.
// Bilinear metadata (weights + clamped gather indices) precomputed per tile,
// matching the reference's pad-then-clip math exactly.  Geometry is fully
// compile-time so all addressing folds to shifts / immediate LDS offsets.
// ---------------------------------------------------------------------------
template <int C, int H, int W, int h, int w, int STRIDE, int Cout>
__global__ __launch_bounds__(128) void deform_gemm_wmma(
    const float* __restrict__ x,    // (B,C,H,W) input features
    const float* __restrict__ off,  // (B,18,h,w) offsets
    const float* __restrict__ wc,   // (Cout,C,3,3)
    const float* __restrict__ g,  const float* __restrict__ be,
    const float* __restrict__ mn, const float* __restrict__ vr,
    float* __restrict__ out)        // (B,Cout,h,w)
{
    constexpr int NT = Cout / 16;
    constexpr int hw = h * w;
    constexpr int Hp = H + 2, Wp = W + 2;
    constexpr int K = C * 9;

    __shared__ float Wgt[4][144][4];   // per-wave bilinear weights (m,n)
    __shared__ int   Idx[4][144][4];   // per-wave gather coords ix0,ix1,iy0,iy1
    __shared__ float At[4][16 * 36];   // per-wave A chunk (M x 36)
    __shared__ float Bt[36 * Cout];    // shared B chunk (36 x Cout)

    const int tid  = threadIdx.x;
    const int wave = tid >> 5;
    const int lane = tid & 31;
    const int mtile = blockIdx.x * 4 + wave;
    const int b    = (mtile * 16) / hw;         // tile never crosses an image
    const int rem0 = (mtile * 16) & (hw - 1);

    // ---- per-tile bilinear metadata (16 pixels x 9 taps) ----
    for (int e = lane; e < 144; e += 32) {
        int m = e / 9, n = e % 9;
        int rem = rem0 + m;
        int y = rem / w, xx = rem & (w - 1);
        const float* ob = off + (long)b * 18 * hw + y * w + xx;
        float ox = ob[n * hw];
        float oy = ob[(n + 9) * hw];
        float px = (float)(1 + y * STRIDE)  + (float)(n / 3 - 1) + ox;
        float py = (float)(1 + xx * STRIDE) + (float)(n % 3 - 1) + oy;
        float qx0 = floorf(px), qy0 = floorf(py);
        float fx0 = fminf(fmaxf(qx0,       0.f), (float)(Hp - 1));
        float fx1 = fminf(fmaxf(qx0 + 1.f, 0.f), (float)(Hp - 1));
        float fy0 = fminf(fmaxf(qy0,       0.f), (float)(Wp - 1));
        float fy1 = fminf(fmaxf(qy0 + 1.f, 0.f), (float)(Wp - 1));
        float pxc = fminf(fmaxf(px, 0.f), (float)(Hp - 1));
        float pyc = fminf(fmaxf(py, 0.f), (float)(Wp - 1));
        float wx0 = 1.f + (fx0 - pxc), wx1 = 1.f - (fx1 - pxc);
        float wy0 = 1.f + (fy0 - pyc), wy1 = 1.f - (fy1 - pyc);
        Wgt[wave][e][0] = wx0 * wy0;   // lt
        Wgt[wave][e][1] = wx1 * wy1;   // rb
        Wgt[wave][e][2] = wx0 * wy1;   // lb
        Wgt[wave][e][3] = wx1 * wy0;   // rt
        Idx[wave][e][0] = (int)fx0 - 1;   // unpadded coords; border -> OOB -> 0
        Idx[wave][e][1] = (int)fx1 - 1;
        Idx[wave][e][2] = (int)fy0 - 1;
        Idx[wave][e][3] = (int)fy1 - 1;
    }

    v8f acc[NT];
#pragma unroll
    for (int i = 0; i < NT; ++i) acc[i] = (v8f){};

    const int mrow = lane & 15;
    const int koff = (lane >> 4) << 1;
    const float* xb0 = x + (long)b * C * H * W;

    for (int cg = 0; cg < C / 4; ++cg) {
        __syncthreads();
        // ---- stage B chunk (whole block) ----
#pragma unroll
        for (int e = tid; e < 36 * Cout; e += 128) {
            int k = e / Cout, o = e & (Cout - 1);
            Bt[e] = wc[o * K + cg * 36 + k];
        }
        // ---- build A chunk via bilinear gathers (per wave) ----
#pragma unroll
        for (int e = lane; e < 16 * 36; e += 32) {
            int m = e / 36, kk = e % 36;
            int c = cg * 4 + kk / 9, n = kk % 9;
            const float* xb = xb0 + c * (H * W);
            int meta = m * 9 + n;
            int ix0 = Idx[wave][meta][0], ix1 = Idx[wave][meta][1];
            int iy0 = Idx[wave][meta][2], iy1 = Idx[wave][meta][3];
            float v00 = (ix0 >= 0 && ix0 < H && iy0 >= 0 && iy0 < W) ? xb[ix0 * W + iy0] : 0.f;
            float v11 = (ix1 >= 0 && ix1 < H && iy1 >= 0 && iy1 < W) ? xb[ix1 * W + iy1] : 0.f;
            float v01 = (ix0 >= 0 && ix0 < H && iy1 >= 0 && iy1 < W) ? xb[ix0 * W + iy1] : 0.f;
            float v10 = (ix1 >= 0 && ix1 < H && iy0 >= 0 && iy0 < W) ? xb[ix1 * W + iy0] : 0.f;
            At[wave][e] = Wgt[wave][meta][0] * v00 + Wgt[wave][meta][1] * v11
                        + Wgt[wave][meta][2] * v01 + Wgt[wave][meta][3] * v10;
        }
        __syncthreads();

        // ---- 9 WMMA k-steps over the 36-wide chunk ----
        const float* Ar = &At[wave][mrow * 36];
#pragma unroll
        for (int s = 0; s < 9; ++s) {
            int kb = s * 4 + koff;
            v2f a; a.x = Ar[kb]; a.y = Ar[kb + 1];
#pragma unroll
            for (int nt = 0; nt < NT; ++nt) {
                int n = mrow + nt * 16;
                v2f bb; bb.x = Bt[kb * Cout + n]; bb.y = Bt[(kb + 1) * Cout + n];
                acc[nt] = __builtin_amdgcn_wmma_f32_16x16x4_f32(
                    false, a, false, bb, (short)0, acc[nt], false, false);
            }
        }
    }

    // ---- epilogue: BN + ReLU, scatter to NCHW ----
    const int mb = (lane >> 4) * 8;
#pragma unroll
    for (int nt = 0; nt < NT; ++nt) {
        int o = mrow + nt * 16;
        float sc = g[o] * rsqrtf(vr[o] + BN_EPS);
        float sh = be[o] - mn[o] * sc;
#pragma unroll
        for (int r = 0; r < 8; ++r) {
            int rem = rem0 + mb + r;
            int y = rem / w, xx = rem & (w - 1);
            float v = acc[nt][r] * sc + sh;
            out[(((long)b * Cout + o) * h + y) * w + xx] = v > 0.f ? v : 0.f;
        }
    }
}

// ---------------------------------------------------------------------------
// Global average pool (B,128,8,8) -> (B,128)
// ---------------------------------------------------------------------------
__global__ void avgpool_k(const float* __restrict__ hin, float* __restrict__ pooled)
{
    int idx = blockIdx.x * blockDim.x + threadIdx.x;   // B*128 threads
    if (idx >= 256 * 128) return;
    const float* p = hin + (long)idx * 64;
    float s = 0.f;
#pragma unroll
    for (int i = 0; i < 64; ++i) s += p[i];
    pooled[idx] = s * (1.0f / 64.0f);
}

// ---------------------------------------------------------------------------
// Classifier: (B,128) @ (100,128)^T + bias -> (B,100)
// ---------------------------------------------------------------------------
__global__ void fc_k(const float* __restrict__ pooled,
                     const float* __restrict__ wcls, const float* __restrict__ bcls,
                     float* __restrict__ out)
{
    int idx = blockIdx.x * blockDim.x + threadIdx.x;   // B*100 threads
    if (idx >= 256 * 100) return;
    int o = idx % 100, b = idx / 100;
    const float* pv = pooled + b * 128;
    const float* wv = wcls + o * 128;
    float acc = bcls[o];
#pragma unroll 8
    for (int k = 0; k < 128; ++k) acc += pv[k] * wv[k];
    out[idx] = acc;
}

// ---------------------------------------------------------------------------
extern "C" void kernel_launch(void* const* d_in, const int* in_sizes, int n_in,
                              void* d_out, int out_size, void* d_ws, size_t ws_size,
                              hipStream_t stream)
{
    const float* x    = (const float*)d_in[0];
    const float* w1   = (const float*)d_in[1];
    const float* b1   = (const float*)d_in[2];
    const float* g1   = (const float*)d_in[3];
    const float* be1  = (const float*)d_in[4];
    const float* m1   = (const float*)d_in[5];
    const float* v1   = (const float*)d_in[6];
    const float* wp2  = (const float*)d_in[7];
    const float* bp2  = (const float*)d_in[8];
    const float* wc2  = (const float*)d_in[9];
    const float* g2   = (const float*)d_in[10];
    const float* be2  = (const float*)d_in[11];
    const float* m2   = (const float*)d_in[12];
    const float* v2   = (const float*)d_in[13];
    const float* wp3  = (const float*)d_in[14];
    const float* bp3  = (const float*)d_in[15];
    const float* wc3  = (const float*)d_in[16];
    const float* g3   = (const float*)d_in[17];
    const float* be3  = (const float*)d_in[18];
    const float* m3   = (const float*)d_in[19];
    const float* v3   = (const float*)d_in[20];
    const float* wp4  = (const float*)d_in[21];
    const float* bp4  = (const float*)d_in[22];
    const float* wc4  = (const float*)d_in[23];
    const float* g4   = (const float*)d_in[24];
    const float* be4  = (const float*)d_in[25];
    const float* m4   = (const float*)d_in[26];
    const float* v4   = (const float*)d_in[27];
    const float* wcls = (const float*)d_in[28];
    const float* bcls = (const float*)d_in[29];
    float* out = (float*)d_out;

    // Workspace layout (floats)
    float* ws = (float*)d_ws;
    float* buf1   = ws;                         // (256,32,32,32)  8388608
    float* off2   = buf1 + 8388608;             // (256,18,16,16)  1179648
    float* buf2   = off2 + 1179648;             // (256,64,16,16)  4194304
    float* off3   = buf2 + 4194304;             // (256,18,16,16)  1179648
    float* buf3   = off3 + 1179648;             // (256,128,16,16) 8388608
    float* off4   = buf3 + 8388608;             // (256,18,8,8)     294912
    float* buf4   = off4 + 294912;              // (256,128,8,8)   2097152
    float* pooled = buf4 + 2097152;             // (256,128)         32768

    // Stage 1: conv1 + BN + ReLU   (M tiles = 262144/16 = 16384; 4 waves/block)
    conv1_wmma<<<4096, 128, 0, stream>>>(x, w1, b1, g1, be1, m1, v1, buf1);

    // Stage 2: deform conv 32->64, stride 2
    offset_conv<32, 32, 32, 16, 16, 2><<<4608, 256, 0, stream>>>(buf1, wp2, bp2, off2);
    deform_gemm_wmma<32, 32, 32, 16, 16, 2, 64>
        <<<1024, 128, 0, stream>>>(buf1, off2, wc2, g2, be2, m2, v2, buf2);

    // Stage 3: deform conv 64->128, stride 1
    offset_conv<64, 16, 16, 16, 16, 1><<<4608, 256, 0, stream>>>(buf2, wp3, bp3, off3);
    deform_gemm_wmma<64, 16, 16, 16, 16, 1, 128>
        <<<1024, 128, 0, stream>>>(buf2, off3, wc3, g3, be3, m3, v3, buf3);

    // Stage 4: deform conv 128->128, stride 2
    offset_conv<128, 16, 16, 8, 8, 2><<<1152, 256, 0, stream>>>(buf3, wp4, bp4, off4);
    deform_gemm_wmma<128, 16, 16, 8, 8, 2, 128>
        <<<256, 128, 0, stream>>>(buf3, off4, wc4, g4, be4, m4, v4, buf4);

    // Pool + classifier
    avgpool_k<<<128, 256, 0, stream>>>(buf4, pooled);
    fc_k<<<100, 256, 0, stream>>>(pooled, wcls, bcls, out);
}